// YoloLayer_51075751084308
// MI455X (gfx1250) — compile-verified
//
#include <hip/hip_runtime.h>

#ifndef __has_builtin
#define __has_builtin(x) 0
#endif

// ---- YOLOv4 stride-8 head constants (fixed by the reference config) ----
#define YG      76                  // grid dim
#define YG2     (YG * YG)           // 5776
#define YA      3                   // anchors in this head
#define YATTRS  85                  // 5 bbox attrs + 80 classes
#define YTILE   (YG * YATTRS)       // 6460 elements per (b,i,a) slab
#define YBLOCK  256                 // 8 wave32
#define YSTRIDE 8.0f                // 608 / 76

// Use the gfx1250 async global->LDS copy engine when the toolchain exposes it.
#if defined(__HIP_DEVICE_COMPILE__) && __has_builtin(__builtin_amdgcn_global_load_async_to_lds_b32)
#define USE_ASYNC_LDS 1
#else
#define USE_ASYNC_LDS 0
#endif

#if USE_ASYNC_LDS
typedef __attribute__((address_space(1))) int* g_i32p;   // global (printed as __device__ int*)
typedef __attribute__((address_space(3))) int* l_i32p;   // LDS
#endif

__global__ __launch_bounds__(YBLOCK)
void yolo_head_s8(const float* __restrict__ x, float* __restrict__ out)
{
    // [j][c] transposed staging tile; 76*85*4 = 25840 B
    __shared__ float tile[YTILE];

    const int bid = blockIdx.x;            // flat over B * G * A
    const int a   = bid % YA;
    const int i   = (bid / YA) % YG;
    const int b   = bid / (YA * YG);

    // input slab base: x[b][a][c][i][j] -> rows (c) are G2 apart, j contiguous
    const float* __restrict__ src =
        x + ((size_t)(b * YA + a) * YATTRS) * YG2 + (size_t)i * YG;
    // output base: out[b][i][j][a][c]; element (j,c) at dst + j*255 + c
    float* __restrict__ dst =
        out + ((size_t)((b * YG + i) * YG) * YA + a) * YATTRS;

    const int t = threadIdx.x;

    // ---------- Phase 1: stage [c][j] slab into LDS transposed as [j][c] ----------
#if USE_ASYNC_LDS
    for (int idx = t; idx < YTILE; idx += YBLOCK) {
        const int c = idx / YG;
        const int j = idx - c * YG;
        // coalesced global read (lanes walk j), scattered per-lane LDS write = transpose
        __builtin_amdgcn_global_load_async_to_lds_b32(
            (g_i32p)const_cast<float*>(src + (size_t)c * YG2 + j),
            (l_i32p)&tile[j * YATTRS + c],
            /*offset=*/0, /*cpol=*/0);
    }
#if __has_builtin(__builtin_amdgcn_s_wait_asynccnt)
    __builtin_amdgcn_s_wait_asynccnt(0);
#else
    asm volatile("s_wait_asynccnt 0x0" ::: "memory");
#endif
#else
    for (int idx = t; idx < YTILE; idx += YBLOCK) {
        const int c = idx / YG;
        const int j = idx - c * YG;
        tile[j * YATTRS + c] = __builtin_nontemporal_load(src + (size_t)c * YG2 + j);
    }
#endif
    __syncthreads();

    // anchors for mask [0,1,2]; exp(t)*(anchor/stride)*stride == exp(t)*anchor
    const float AW[YA] = {12.f, 19.f, 40.f};
    const float AH[YA] = {16.f, 36.f, 28.f};
    const float aw = AW[a], ah = AH[a];
    const float fi = (float)i;

    // ---------- Phase 2: elementwise transform + coalesced store ----------
    for (int idx = t; idx < YTILE; idx += YBLOCK) {
        const int j = idx / YATTRS;
        const int c = idx - j * YATTRS;
        const float v = tile[idx];          // contiguous, conflict-free LDS read

        const bool isWH = (c == 2) || (c == 3);
        const float e   = __expf(isWH ? v : -v);          // single v_exp_f32
        const float sig = __builtin_amdgcn_rcpf(1.0f + e); // sigmoid via v_rcp_f32

        float r;
        if (c == 0)       r = (sig * 1.2f - 0.1f + (float)j) * YSTRIDE; // tx + gx
        else if (c == 1)  r = (sig * 1.2f - 0.1f + fi) * YSTRIDE;       // ty + gy
        else if (c == 2)  r = e * aw;                                   // tw
        else if (c == 3)  r = e * ah;                                   // th
        else              r = sig;                                      // obj/cls

        // runs of 85 contiguous floats per j; lanes walk c -> coalesced
        __builtin_nontemporal_store(r, dst + (size_t)j * (YA * YATTRS) + c);
    }
}

extern "C" void kernel_launch(void* const* d_in, const int* in_sizes, int n_in,
                              void* d_out, int out_size, void* d_ws, size_t ws_size,
                              hipStream_t stream)
{
    (void)n_in; (void)out_size; (void)d_ws; (void)ws_size;
    const float* x = (const float*)d_in[0];     // [B, 255, 76, 76] fp32
    float* out = (float*)d_out;                 // [B, 76*76*3, 85] fp32
    const int B = in_sizes[0] / (YA * YATTRS * YG2);   // 16 for the reference shapes
    dim3 grid((unsigned)(B * YG * YA));                // 3648 blocks
    yolo_head_s8<<<grid, YBLOCK, 0, stream>>>(x, out);
}